// ElementCWLinear_17600775979358
// MI455X (gfx1250) — compile-verified
//
#include <hip/hip_runtime.h>
#include <stdint.h>

// ---------------------------------------------------------------------------
// out[b,c,d] = sum_p (ALPHA * W[e_b, c, p]) * t[p,b,c,d]
//   e_b = index of the 1.0 in one-hot node_attrs[b,:]
// Memory-bound streaming kernel: ~780 MB moved, ~0.3 GFLOP -> HBM roofline
// (~34 us at 23.3 TB/s). WMMA structurally inapplicable (one-hot == gather,
// contraction length P=4 with per-(b,c) weights). The CDNA5 path that matters
// is data movement: global_load_async_to_lds_b128 (ASYNCcnt DMA) with
// double-buffered LDS tiles and s_wait_asynccnt pipelining.
// ---------------------------------------------------------------------------

#define Pn     4
#define Bn     50000
#define Cn     256
#define Dn     3
#define Zn     64
#define NB     8                 // nodes per workgroup
#define NT     256               // threads per workgroup (8 wave32)
#define CHUNK  (Cn * Dn)         // 768 floats per (p, node) chunk = 3 KB
#define NODE_F (Pn * CHUNK)      // 3072 floats per node tile = 12 KB
#define ALPHA  0.5f              // 1/sqrt(P)

#if defined(__has_builtin)
#  if __has_builtin(__builtin_amdgcn_global_load_async_to_lds_b128) && \
      __has_builtin(__builtin_amdgcn_s_wait_asynccnt)
#    define USE_ASYNC 1
#  endif
#endif
#ifndef USE_ASYNC
#  define USE_ASYNC 0
#endif

#if USE_ASYNC
#  define WAIT_ASYNC(n) __builtin_amdgcn_s_wait_asynccnt(n)
#else
#  define WAIT_ASYNC(n) ((void)0)
#endif

// Exact pointee type from the builtin's signature: v4i in as(1)/as(3).
typedef int v4i __attribute__((ext_vector_type(4)));
typedef __attribute__((address_space(1))) v4i* gv4i_p;
typedef __attribute__((address_space(3))) v4i* lv4i_p;

// Stage one node's 12 KB t-tile into an LDS buffer.
// 4 chunks of 3 KB; threads 0..191 each move 16 B per chunk (b128).
__device__ __forceinline__ void stage_node(float* __restrict__ dst,
                                           const float* __restrict__ t,
                                           int b, int tid) {
    if (tid < CHUNK / 4) {  // 192 lanes x 16 B = 3 KB per chunk
#pragma unroll
        for (int p = 0; p < Pn; ++p) {
            const float* g = t + ((size_t)p * Bn + (size_t)b) * CHUNK + tid * 4;
            float*       l = dst + p * CHUNK + tid * 4;
#if USE_ASYNC
            __builtin_amdgcn_global_load_async_to_lds_b128(
                (gv4i_p)(uintptr_t)g,
                (lv4i_p)(uint32_t)(uintptr_t)l,
                /*imm offset*/ 0, /*cpol*/ 0);
#else
            *(float4*)l = *(const float4*)g;
#endif
        }
    }
}

__global__ __launch_bounds__(NT) void
ElementCWLinear_17600775979358_kernel(const float* __restrict__ t,
                                      const float* __restrict__ node_attrs,
                                      const float* __restrict__ weights,
                                      float* __restrict__ out) {
    __shared__ float lt[2][NODE_F];   // double-buffered t tiles (24 KB)
    __shared__ int   s_e[NB];         // element index per node

    const int tid = threadIdx.x;
    const int b0  = blockIdx.x * NB;

    // Recover element indices from the one-hot rows (exactly one hit each).
    // NB*Zn == 512 == 2*NT: two constant-stride steps, no runtime division.
#pragma unroll
    for (int k = 0; k < (NB * Zn) / NT; ++k) {
        const int idx = tid + k * NT;
        const int n   = idx >> 6;          // node within block
        const int z   = idx & (Zn - 1);    // atomic-number slot
        if (node_attrs[(size_t)(b0 + n) * Zn + z] > 0.5f) s_e[n] = z;
    }

    // Kick off DMA for node 0 while indices settle.
    stage_node(lt[0], t, b0, tid);
    __syncthreads();   // s_e[] visible to all waves

#pragma unroll 1
    for (int i = 0; i < NB; ++i) {
        const int cur = i & 1;

        // Prefetch next node's tile into the other buffer.
        if (i + 1 < NB) {
            stage_node(lt[cur ^ 1], t, b0 + i + 1, tid);
            WAIT_ASYNC(4);   // async loads retire in order: current tile done
        } else {
            WAIT_ASYNC(0);   // drain
        }
        __syncthreads();     // cross-wave visibility of LDS tile

        // ---- compute node b0+i: thread tid owns channel c = tid ----
        const int b = b0 + i;
        const int e = s_e[i];

        float4 w = *(const float4*)(weights + ((size_t)e * Cn + tid) * Pn);
        w.x *= ALPHA; w.y *= ALPHA; w.z *= ALPHA; w.w *= ALPHA;

        const float* base = &lt[cur][tid * Dn];
        const float  a0 = w.x * base[0]
                        + w.y * base[CHUNK + 0]
                        + w.z * base[2 * CHUNK + 0]
                        + w.w * base[3 * CHUNK + 0];
        const float  a1 = w.x * base[1]
                        + w.y * base[CHUNK + 1]
                        + w.z * base[2 * CHUNK + 1]
                        + w.w * base[3 * CHUNK + 1];
        const float  a2 = w.x * base[2]
                        + w.y * base[CHUNK + 2]
                        + w.z * base[2 * CHUNK + 2]
                        + w.w * base[3 * CHUNK + 2];

        float* o = out + ((size_t)b * Cn + tid) * Dn;
        o[0] = a0; o[1] = a1; o[2] = a2;

        __syncthreads();     // all waves done reading lt[cur] before reuse
    }
}

extern "C" void kernel_launch(void* const* d_in, const int* in_sizes, int n_in,
                              void* d_out, int out_size, void* d_ws, size_t ws_size,
                              hipStream_t stream) {
    (void)in_sizes; (void)n_in; (void)out_size; (void)d_ws; (void)ws_size;
    const float* t          = (const float*)d_in[0];  // [P,B,C,3]
    const float* node_attrs = (const float*)d_in[1];  // [B,Z] one-hot
    const float* weights    = (const float*)d_in[2];  // [Z,C,P]
    float*       out        = (float*)d_out;          // [B,C,3]

    ElementCWLinear_17600775979358_kernel<<<Bn / NB, NT, 0, stream>>>(
        t, node_attrs, weights, out);
}